// TokenSparse_730144441133
// MI455X (gfx1250) — compile-verified
//
#include <hip/hip_runtime.h>
#include <stdint.h>

#define B_ 64
#define L_ 4096
#define C_ 512
#define K_ 2458          // ceil(4096 * 0.6)
#define T_ 1638          // L_ - K_
#define CHUNK_ 64
#define NCH_ 26          // ceil(T_ / CHUNK_)

// ---- monotonic float <-> uint key helpers (descending-stable sort) ----
__device__ __forceinline__ unsigned fkey_asc(float f) {
    unsigned b = __float_as_uint(f);
    return (b & 0x80000000u) ? ~b : (b | 0x80000000u);   // ascending-monotonic
}
__device__ __forceinline__ float fkey_decode_desc(unsigned kd) {
    unsigned u = ~kd;                                    // undo descending flip
    unsigned b = (u & 0x80000000u) ? (u ^ 0x80000000u) : ~u;
    return __uint_as_float(b);
}

// ============================================================================
// Kernel 1: per-batch score, bitonic argsort (stable, descending), mask,
//           tail softmax weights. One 1024-thread workgroup (32 waves) / batch.
// 32 KB u64 keys live in the 320 KB WGP LDS.
// ============================================================================
__global__ __launch_bounds__(1024) void sort_score_kernel(
    const float* __restrict__ ax, const float* __restrict__ ay,
    int* __restrict__ keep_idx, int* __restrict__ tail_idx,
    float* __restrict__ tail_w, float* __restrict__ mask_out)
{
    __shared__ unsigned long long keys[L_];
    __shared__ float red[1024];

    const int b    = blockIdx.x;
    const int tid  = threadIdx.x;
    const int nthr = blockDim.x;   // 1024

    // keys: high 32 = ~monotonic(score) => ascending u64 sort == descending
    // score; low 32 = original index => stable tie-break (matches jnp.argsort)
    for (int i = tid; i < L_; i += nthr) {
        float s = ax[b * L_ + i] + ay[b * L_ + i];
        unsigned kd = ~fkey_asc(s);
        keys[i] = ((unsigned long long)kd << 32) | (unsigned)i;
    }
    __syncthreads();

    // bitonic sort, ascending on u64 keys
    for (unsigned k = 2; k <= (unsigned)L_; k <<= 1) {
        for (unsigned j = k >> 1; j > 0; j >>= 1) {
            for (unsigned i = tid; i < (unsigned)L_; i += nthr) {
                unsigned ixj = i ^ j;
                if (ixj > i) {
                    unsigned long long a = keys[i];
                    unsigned long long c = keys[ixj];
                    bool up = ((i & k) == 0);
                    if ((a > c) == up) { keys[i] = c; keys[ixj] = a; }
                }
            }
            __syncthreads();
        }
    }

    // emit keep/tail index lists + 0/1 mask scatter (covers every position)
    for (int i = tid; i < L_; i += nthr) {
        unsigned long long kv = keys[i];
        int idx = (int)(unsigned)kv;
        mask_out[b * L_ + idx] = (i < K_) ? 1.0f : 0.0f;
        if (i < K_) keep_idx[b * K_ + i]        = idx;
        else        tail_idx[b * T_ + (i - K_)] = idx;
    }
    __syncthreads();

    // softmax over tail scores (sorted descending => max is keys[K_])
    float smax = fkey_decode_desc((unsigned)(keys[K_] >> 32));
    float part = 0.0f;
    for (int i = K_ + tid; i < L_; i += nthr)
        part += __expf(fkey_decode_desc((unsigned)(keys[i] >> 32)) - smax);
    red[tid] = part;
    __syncthreads();
    for (int s = nthr >> 1; s > 0; s >>= 1) {
        if (tid < s) red[tid] += red[tid + s];
        __syncthreads();
    }
    float inv = 1.0f / red[0];
    for (int i = K_ + tid; i < L_; i += nthr) {
        float w = __expf(fkey_decode_desc((unsigned)(keys[i] >> 32)) - smax) * inv;
        tail_w[b * T_ + (i - K_)] = w;
    }
}

// ============================================================================
// Kernel 2: gather kept token rows via the CDNA5 async global<->LDS DMA path
// (ASYNCcnt). Each thread moves TWO 16B chunks (two rows per 128-lane group):
// both async b128 loads are issued before a single s_wait_asynccnt 0, doubling
// bytes in flight per wave, then both b128 stores go out. All chunks are
// thread-private, so no barriers are needed. 8 rows (16 KB) per 512-thread
// block; B_*K_ = 157312 is divisible by 8 so no partial blocks.
// ============================================================================
__global__ __launch_bounds__(512) void gather_tokens_kernel(
    const float* __restrict__ tokens, const int* __restrict__ keep_idx,
    float* __restrict__ out_select)
{
    __shared__ float buf[8 * C_];   // 16 KB bounce buffer
    const int tid  = threadIdx.x;
    const int lane = tid & 127;     // 16B chunk within row
    const int rsub = tid >> 7;      // 0..3

    const long long row0 = (long long)blockIdx.x * 8 + rsub;      // first row
    const long long row1 = row0 + 4;                              // second row

    // row0 source/dest
    int b0 = (int)(row0 / K_);
    int i0 = (int)(row0 % K_);
    int src_r0 = keep_idx[b0 * K_ + i0];
    unsigned long long gsrc0 = (unsigned long long)(size_t)
        (tokens + ((size_t)b0 * L_ + src_r0) * C_ + lane * 4);
    unsigned long long gdst0 = (unsigned long long)(size_t)
        (out_select + (size_t)row0 * C_ + lane * 4);
    unsigned lds0 = (unsigned)(size_t)(&buf[rsub * C_ + lane * 4]);

    // row1 source/dest
    int b1 = (int)(row1 / K_);
    int i1 = (int)(row1 % K_);
    int src_r1 = keep_idx[b1 * K_ + i1];
    unsigned long long gsrc1 = (unsigned long long)(size_t)
        (tokens + ((size_t)b1 * L_ + src_r1) * C_ + lane * 4);
    unsigned long long gdst1 = (unsigned long long)(size_t)
        (out_select + (size_t)row1 * C_ + lane * 4);
    unsigned lds1 = (unsigned)(size_t)(&buf[(rsub + 4) * C_ + lane * 4]);

    asm volatile("global_load_async_to_lds_b128 %0, %1, off"
                 :: "v"(lds0), "v"(gsrc0) : "memory");
    asm volatile("global_load_async_to_lds_b128 %0, %1, off"
                 :: "v"(lds1), "v"(gsrc1) : "memory");
    asm volatile("s_wait_asynccnt 0" ::: "memory");
    asm volatile("global_store_async_from_lds_b128 %0, %1, off"
                 :: "v"(gdst0), "v"(lds0) : "memory");
    asm volatile("global_store_async_from_lds_b128 %0, %1, off"
                 :: "v"(gdst1), "v"(lds1) : "memory");
    // s_endpgm performs an implicit wait-idle, draining ASYNCcnt
}

// ============================================================================
// Kernel 3: partial weighted sums of tail tokens (deterministic, no atomics).
// grid = (B, NCH), block = C threads (one channel per thread, coalesced).
// ============================================================================
__global__ __launch_bounds__(C_) void extra_partial_kernel(
    const float* __restrict__ tokens, const int* __restrict__ tail_idx,
    const float* __restrict__ tail_w, float* __restrict__ partials)
{
    __shared__ int   sidx[CHUNK_];
    __shared__ float sw[CHUNK_];
    const int b   = blockIdx.x;
    const int cn  = blockIdx.y;
    const int tid = threadIdx.x;

    int t0 = cn * CHUNK_;
    int tn = min(CHUNK_, T_ - t0);
    if (tid < tn) {
        sidx[tid] = tail_idx[b * T_ + t0 + tid];
        sw[tid]   = tail_w[b * T_ + t0 + tid];
    }
    __syncthreads();

    float acc = 0.0f;
    for (int t = 0; t < tn; ++t)
        acc += sw[t] * tokens[((size_t)b * L_ + sidx[t]) * C_ + tid];

    partials[((size_t)b * NCH_ + cn) * C_ + tid] = acc;
}

// ============================================================================
// Kernel 4: reduce partials -> extra_token (fixed order => deterministic)
// ============================================================================
__global__ __launch_bounds__(C_) void extra_reduce_kernel(
    const float* __restrict__ partials, float* __restrict__ out_extra)
{
    const int b   = blockIdx.x;
    const int tid = threadIdx.x;
    float acc = 0.0f;
    for (int cn = 0; cn < NCH_; ++cn)
        acc += partials[((size_t)b * NCH_ + cn) * C_ + tid];
    out_extra[b * C_ + tid] = acc;
}

// ============================================================================
extern "C" void kernel_launch(void* const* d_in, const int* in_sizes, int n_in,
                              void* d_out, int out_size, void* d_ws, size_t ws_size,
                              hipStream_t stream) {
    const float* tokens = (const float*)d_in[0];
    const float* ax     = (const float*)d_in[1];
    const float* ay     = (const float*)d_in[2];

    float* out        = (float*)d_out;
    float* out_select = out;                               // B*K*C
    float* out_extra  = out + (size_t)B_ * K_ * C_;        // B*C
    float* out_mask   = out_extra + (size_t)B_ * C_;       // B*L

    char* ws = (char*)d_ws;
    int*   keep_idx = (int*)ws;    ws += (size_t)B_ * K_ * sizeof(int);
    int*   tail_idx = (int*)ws;    ws += (size_t)B_ * T_ * sizeof(int);
    float* tail_w   = (float*)ws;  ws += (size_t)B_ * T_ * sizeof(float);
    float* partials = (float*)ws;  // B * NCH_ * C_ floats (~3.4 MB)

    sort_score_kernel<<<B_, 1024, 0, stream>>>(ax, ay, keep_idx, tail_idx,
                                               tail_w, out_mask);
    gather_tokens_kernel<<<(B_ * K_) / 8, 512, 0, stream>>>(tokens, keep_idx,
                                                            out_select);
    extra_partial_kernel<<<dim3(B_, NCH_), C_, 0, stream>>>(tokens, tail_idx,
                                                            tail_w, partials);
    extra_reduce_kernel<<<B_, C_, 0, stream>>>(partials, out_extra);
}